// RBFLayer_8881992368406
// MI455X (gfx1250) — compile-verified
//
#include <hip/hip_runtime.h>
#include <hip/hip_bf16.h>

typedef float v2f __attribute__((ext_vector_type(2)));
typedef float v8f __attribute__((ext_vector_type(8)));

// Tiling constants
static constexpr int K_FEAT     = 64;   // features
static constexpr int ROWS_WG    = 16;   // rows (x) per workgroup
static constexpr int WAVES      = 8;    // waves per workgroup (wave32)
static constexpr int COL_ITERS  = 2;    // 16-col tiles per wave
static constexpr int COLS_WG    = WAVES * 16 * COL_ITERS; // 256 cols per WG

__global__ __launch_bounds__(256)
void rbf_wmma_f32_kernel(const float* __restrict__ x,
                         const float* __restrict__ centers,
                         float* __restrict__ out,
                         int U /* num centers = 2048 */)
{
    const int lane = threadIdx.x & 31;
    const int wave = threadIdx.x >> 5;
    const int m = lane & 15;   // row/col within 16-tile
    const int h = lane >> 4;   // lane half (K split for A/B, M split for C/D)

    const int rowbase  = blockIdx.y * ROWS_WG;
    const int colstrip = blockIdx.x * COLS_WG;

    // ---- Load A tile: lane holds features {4s+2h, 4s+2h+1} of x row (rowbase+m)
    // This matches V_WMMA_F32_16X16X4_F32 A layout (16x4, VGPR0=K0/K2, VGPR1=K1/K3).
    const float* __restrict__ xrow = x + (size_t)(rowbase + m) * K_FEAT;
    v2f a[16];
    float half_xsq = 0.0f;
#pragma unroll
    for (int s = 0; s < 16; ++s) {
        a[s] = *(const v2f*)(xrow + 4 * s + 2 * h);
        half_xsq += a[s].x * a[s].x + a[s].y * a[s].y;
    }
    // Lane l and l^16 hold complementary feature halves of the same row.
    const float full_xsq = half_xsq + __shfl_xor(half_xsq, 16);
    // C/D element r lives at M = r + 8*h; fetch that row's norm.
    float rowsq[8];
#pragma unroll
    for (int r = 0; r < 8; ++r)
        rowsq[r] = __shfl(full_xsq, r + 8 * h);

    const float gamma = 1.0f / (float)K_FEAT;

#pragma unroll
    for (int it = 0; it < COL_ITERS; ++it) {
        const int colbase = colstrip + wave * 16 + it * (WAVES * 16);

        // ---- Load B tile: lane holds features {4s+2h, 4s+2h+1} of center (colbase+m)
        const float* __restrict__ crow = centers + (size_t)(colbase + m) * K_FEAT;
        v2f b[16];
        float half_csq = 0.0f;
#pragma unroll
        for (int s = 0; s < 16; ++s) {
            b[s] = *(const v2f*)(crow + 4 * s + 2 * h);
            half_csq += b[s].x * b[s].x + b[s].y * b[s].y;
        }
        const float csq = half_csq + __shfl_xor(half_csq, 16);

        // ---- 16 chained FP32 WMMA steps accumulate the 16x16 cross tile
        v8f acc = {};
#pragma unroll
        for (int s = 0; s < 16; ++s) {
            acc = __builtin_amdgcn_wmma_f32_16x16x4_f32(
                /*neg_a=*/false, a[s],
                /*neg_b=*/false, b[s],
                /*c_mod=*/(short)0, acc,
                /*reuse_a=*/false, /*reuse_b=*/false);
        }

        // ---- Fused epilogue: l2 = |x|^2 + |c|^2 - 2*cross, out = exp(-gamma*l2)
        // D element r -> out[rowbase + r + 8h][colbase + m]; stream store (NT),
        // output (512MB) exceeds L2 so keep it out of the cache hierarchy.
#pragma unroll
        for (int r = 0; r < 8; ++r) {
            float l2 = rowsq[r] + csq - 2.0f * acc[r];
            l2 = fmaxf(l2, 0.0f);
            const float v = __expf(-gamma * l2);
            float* dst = out + (size_t)(rowbase + r + 8 * h) * U + (colbase + m);
            __builtin_nontemporal_store(v, dst);
        }
    }
}

extern "C" void kernel_launch(void* const* d_in, const int* in_sizes, int n_in,
                              void* d_out, int out_size, void* d_ws, size_t ws_size,
                              hipStream_t stream) {
    const float* x       = (const float*)d_in[0];   // [B, 64] f32
    const float* centers = (const float*)d_in[1];   // [U, 64] f32
    float* out           = (float*)d_out;           // [B, U] f32

    const int B = in_sizes[0] / K_FEAT;   // 65536
    const int U = in_sizes[1] / K_FEAT;   // 2048

    dim3 grid(U / COLS_WG, B / ROWS_WG);  // (8, 4096)
    dim3 block(256);                      // 8 waves (wave32)
    rbf_wmma_f32_kernel<<<grid, block, 0, stream>>>(x, centers, out, U);
}